// BayesianDTW_86397562127158
// MI455X (gfx1250) — compile-verified
//
#include <hip/hip_runtime.h>
#include <stdint.h>

// Bayesian DTW forward, computed in the log2 domain with hardware
// v_exp_f32 / v_log_f32 (TRANS32 ops, co-execute with VALU on CDNA5):
//   logaddexp(x,y) = ln2 * [ max(x',y') + log2(1 + 2^(-|x'-y'|)) ],  x' = x*log2e
// Row recurrence m_j = logaddexp(m_{j-1}+a_j, b_j) solved as a pair-scan:
//   combine((a1,b1),(a2,b2)) = (a1+a2, lae2(b1+a2, b2))
// wave32 shuffle scan + redundant per-wave scan of the 32 wave totals.
// W rows are streamed into LDS with global_load_async_to_lds_b32 (ASYNCcnt),
// double-buffered so row i+1 DMA overlaps row i compute.

#define NEGV  (-1e20f)                       // natural-log domain boundary
#define NEG2  (-1.4426950408889634e20f)      // NEGV * log2(e)
#define LOG2E (1.4426950408889634f)
#define LN2   (0.6931471805599453f)

constexpr int BATCH = 32;
constexpr int NA = 512;
constexpr int NB = 1024;

// logaddexp in base-2 domain: max + log2(1 + exp2(-|x-y|))
__device__ __forceinline__ float lae2(float x, float y) {
    float mx = fmaxf(x, y);
    float t  = __builtin_amdgcn_exp2f(-fabsf(x - y));   // v_exp_f32
    return mx + __builtin_amdgcn_logf(1.0f + t);        // v_log_f32 (log2)
}

__global__ __launch_bounds__(NB) void bdtw_rowscan(const float* __restrict__ W,
                                                   float* __restrict__ mu) {
    __shared__ float prev[NB + 1];   // previous mu row, base-2 domain
    __shared__ float wbuf[2][NB];    // double-buffered W row (natural domain)
    __shared__ float sA[32];
    __shared__ float sB[32];

    const int b    = blockIdx.x;
    const int j    = threadIdx.x;
    const int lane = j & 31;
    const int wid  = j >> 5;

    const float* Wb  = W  + (size_t)b * NA * NB;
    float*       mub = mu + (size_t)b * (NA + 1) * (NB + 1);

    // mu row 0 (natural domain) and LDS prev row (base-2 domain)
    prev[j + 1] = NEG2;
    mub[j + 1]  = NEGV;
    if (j == 0) { prev[0] = 0.0f; mub[0] = 0.0f; }

    // LDS byte offsets (low 32 bits of generic pointer to LDS == LDS offset)
    const unsigned lds0 = (unsigned)(uintptr_t)&wbuf[0][j];
    const unsigned lds1 = (unsigned)(uintptr_t)&wbuf[1][j];
    const unsigned long long wbase = (unsigned long long)(uintptr_t)Wb;

    // async DMA prefetch of W row 0 into wbuf[0]
    asm volatile("global_load_async_to_lds_b32 %0, %1, %2"
                 :: "v"(lds0), "v"(j * 4u), "s"(wbase) : "memory");

    for (int i = 0; i < NA; ++i) {
        // wave's own row-i async copy done, then make all waves' slices visible
        asm volatile("s_wait_asynccnt 0x0" ::: "memory");
        __syncthreads();                                   // barrier 1

        // kick off row i+1 DMA into the other buffer (overlaps the scan)
        if (i + 1 < NA) {
            unsigned dst = ((i + 1) & 1) ? lds1 : lds0;
            unsigned off = (unsigned)((i + 1) * NB + j) * 4u;
            asm volatile("global_load_async_to_lds_b32 %0, %1, %2"
                         :: "v"(dst), "v"(off), "s"(wbase) : "memory");
        }

        float w   = wbuf[i & 1][j] * LOG2E;  // into base-2 domain
        float pj  = prev[j];                 // top-left (mu[i-1, j])
        float pj1 = prev[j + 1];             // top      (mu[i-1, j+1])

        float A  = w;
        float Bv = lae2(pj1, pj) + w;

        // wave32 inclusive shuffle scan of (A,B)
        #pragma unroll
        for (int d = 1; d < 32; d <<= 1) {
            float pa = __shfl_up(A,  d, 32);
            float pb = __shfl_up(Bv, d, 32);
            if (lane >= d) { Bv = lae2(pb + A, Bv); A = pa + A; }
        }
        if (lane == 31) { sA[wid] = A; sB[wid] = Bv; }
        __syncthreads();                                   // barrier 2

        // every wave redundantly scans the 32 wave totals (no extra barrier)
        float a  = sA[lane];
        float bb = sB[lane];
        #pragma unroll
        for (int d = 1; d < 32; d <<= 1) {
            float pa = __shfl_up(a,  d, 32);
            float pb = __shfl_up(bb, d, 32);
            if (lane >= d) { bb = lae2(pb + a, bb); a = pa + a; }
        }
        // exclusive prefix for this wave, broadcast from lane wid-1
        float pA = __shfl(a,  wid - 1, 32);
        float pB = __shfl(bb, wid - 1, 32);
        if (wid == 0) { pA = 0.0f; pB = NEG2; }

        float Bf = lae2(pB + A, Bv);
        float Af = pA + A;
        float m2 = lae2(NEG2 + Af, Bf);      // fold NEG row-carry (as reference)

        prev[j + 1] = m2;                    // next row's "top", base-2 domain
        float* row = mub + (size_t)(i + 1) * (NB + 1);
        row[j + 1] = m2 * LN2;               // back to natural log for output
        if (j == 0) { prev[0] = NEG2; row[0] = NEGV; }
        // next iteration's barrier 1 orders these LDS writes before reuse
    }
}

extern "C" void kernel_launch(void* const* d_in, const int* in_sizes, int n_in,
                              void* d_out, int out_size, void* d_ws, size_t ws_size,
                              hipStream_t stream) {
    const float* W  = (const float*)d_in[0];   // [32, 512, 1024] f32
    float*       mu = (float*)d_out;           // [32, 513, 1025] f32
    (void)in_sizes; (void)n_in; (void)out_size; (void)d_ws; (void)ws_size;
    bdtw_rowscan<<<dim3(BATCH), dim3(NB), 0, stream>>>(W, mu);
}